// _ProbSparseAttention_40922448396434
// MI455X (gfx1250) — compile-verified
//
#include <hip/hip_runtime.h>
#include <float.h>

typedef __attribute__((ext_vector_type(2))) float v2f;
typedef __attribute__((ext_vector_type(4))) float v4f;
typedef __attribute__((ext_vector_type(8))) float v8f;

#define MAXU 48   // u = 45 at this problem size
#define TKF 32    // key tile for flash kernel
#define ASTRIDE 36  // padded A row stride (floats): 16B-aligned, kills 8-way bank conflict

// ---------------------------------------------------------------------------
// WMMA f32 GEMM: C[M,N] = A[M,K] @ B[K,N]   (M%128==0, N%64==0, K%32==0)
// Workgroup = 256 threads = 8 waves arranged 4x2; block tile 128x64.
// Each wave computes a 32x32 tile = 2x2 independent v8f accumulators
// -> 4-way WMMA ILP and 2x LDS fragment reuse per kk-step.
// As: row stride padded to 36 floats (bank-conflict-free half-wave reads).
// Bs: k-pair-packed layout Bs[k/2][64][2] so each B fragment is one b64 read
//     (no VGPR packing moves before the WMMAs).
// Global loads for the next k-chunk are register-prefetched (b128) while the
// current chunk's 32 WMMAs execute.
// head_major: scatter output into (B,H,L,D) layout instead of (M,N).
// ---------------------------------------------------------------------------
__global__ __launch_bounds__(256) void gemm_wmma(const float* __restrict__ A,
                                                 const float* __restrict__ B,
                                                 float* __restrict__ C,
                                                 int M, int N, int K,
                                                 int head_major) {
    constexpr int Lc = 8192, Hc = 8, Dc = 64;   // fixed problem geometry
    __shared__ float As[128 * ASTRIDE];   // 18 KB
    __shared__ float Bs[32 * 64];         //  8 KB (pair-packed: [k/2][64][2])
    const int tid  = threadIdx.x;
    const int lane = tid & 31;
    const int wid  = tid >> 5;
    const int mb   = (wid & 3) * 32;   // wave row base within block tile
    const int nb   = (wid >> 2) * 32;  // wave col base within block tile
    const int m0   = blockIdx.y * 128;
    const int n0   = blockIdx.x * 64;

    v8f acc00 = {}, acc01 = {}, acc10 = {}, acc11 = {};
    v4f ra[4], rb[2];

    // prologue: prefetch k-chunk 0 into registers
#pragma unroll
    for (int i = 0; i < 4; ++i) {
        int e4 = tid + (i << 8);
        ra[i] = *(const v4f*)&A[(size_t)(m0 + (e4 >> 3)) * K + ((e4 & 7) << 2)];
    }
#pragma unroll
    for (int i = 0; i < 2; ++i) {
        int e4 = tid + (i << 8);
        rb[i] = *(const v4f*)&B[(size_t)(e4 >> 4) * N + n0 + ((e4 & 15) << 2)];
    }

    for (int k0 = 0; k0 < K; k0 += 32) {
        __syncthreads();   // previous chunk's compute done -> safe to overwrite LDS
#pragma unroll
        for (int i = 0; i < 4; ++i) {
            int e4  = tid + (i << 8);
            int row = e4 >> 3, c4 = (e4 & 7) << 2;
            *(v4f*)&As[row * ASTRIDE + c4] = ra[i];
        }
#pragma unroll
        for (int i = 0; i < 2; ++i) {
            int e4 = tid + (i << 8);
            int r  = e4 >> 4, c4 = (e4 & 15) << 2;
            int base = (r >> 1) * 128 + (r & 1);   // pair-packed scatter
            v4f v = rb[i];
            Bs[base + (c4 + 0) * 2] = v.x;
            Bs[base + (c4 + 1) * 2] = v.y;
            Bs[base + (c4 + 2) * 2] = v.z;
            Bs[base + (c4 + 3) * 2] = v.w;
        }
        __syncthreads();

        // register-prefetch next k-chunk while WMMAs run
        int kn = k0 + 32;
        if (kn < K) {
#pragma unroll
            for (int i = 0; i < 4; ++i) {
                int e4 = tid + (i << 8);
                ra[i] = *(const v4f*)&A[(size_t)(m0 + (e4 >> 3)) * K + kn + ((e4 & 7) << 2)];
            }
#pragma unroll
            for (int i = 0; i < 2; ++i) {
                int e4 = tid + (i << 8);
                rb[i] = *(const v4f*)&B[(size_t)(kn + (e4 >> 4)) * N + n0 + ((e4 & 15) << 2)];
            }
        }

#pragma unroll
        for (int kk = 0; kk < 32; kk += 4) {
            // A frags (16x4 f32): lanes 0-15 hold K=kk,kk+1; lanes 16-31 K=kk+2,kk+3
            int kb  = kk + ((lane >> 4) << 1);
            int am0 = mb + (lane & 15);
            v2f a0 = *(const v2f*)&As[am0 * ASTRIDE + kb];
            v2f a1 = *(const v2f*)&As[(am0 + 16) * ASTRIDE + kb];
            // B frags: pair-packed -> single b64 per fragment, no packing moves
            int kp  = kb >> 1;
            int bn0 = nb + (lane & 15);
            v2f b0 = *(const v2f*)&Bs[kp * 128 + bn0 * 2];
            v2f b1 = *(const v2f*)&Bs[kp * 128 + (bn0 + 16) * 2];
            acc00 = __builtin_amdgcn_wmma_f32_16x16x4_f32(false, a0, false, b0, (short)0, acc00, false, false);
            acc01 = __builtin_amdgcn_wmma_f32_16x16x4_f32(false, a0, false, b1, (short)0, acc01, false, false);
            acc10 = __builtin_amdgcn_wmma_f32_16x16x4_f32(false, a1, false, b0, (short)0, acc10, false, false);
            acc11 = __builtin_amdgcn_wmma_f32_16x16x4_f32(false, a1, false, b1, (short)0, acc11, false, false);
        }
    }

    // C/D layout: VGPR i -> row i (lanes 0-15) / row i+8 (lanes 16-31), col = lane&15
    v8f accs[2][2] = {{acc00, acc01}, {acc10, acc11}};
#pragma unroll
    for (int tm = 0; tm < 2; ++tm) {
#pragma unroll
        for (int tn = 0; tn < 2; ++tn) {
            v8f acc = accs[tm][tn];
#pragma unroll
            for (int i = 0; i < 8; ++i) {
                int row = i + ((lane >> 4) << 3);
                int col = lane & 15;
                int gm  = m0 + mb + tm * 16 + row;
                int gn  = n0 + nb + tn * 16 + col;
                float v = acc[i];
                if (head_major) {
                    int b = gm / Lc, l = gm % Lc;   // strength-reduced (pow2)
                    int h = gn / Dc, d = gn % Dc;
                    C[(((size_t)b * Hc + h) * Lc + l) * Dc + d] = v;
                } else {
                    C[(size_t)gm * N + gn] = v;
                }
            }
        }
    }
}

// ---------------------------------------------------------------------------
// M[bh,l] = max_s(scale*q_l.k_s) - mean_s(scale*q_l.k_s) over u sampled keys
// ---------------------------------------------------------------------------
__global__ __launch_bounds__(256) void sample_m_kernel(const float* __restrict__ Q,
                                                       const float* __restrict__ K,
                                                       const int* __restrict__ idx_sample,
                                                       float* __restrict__ Msc,
                                                       int u, int L) {
    __shared__ float Ksamp[MAXU * 64];
    const int tid = threadIdx.x;
    const int bh  = blockIdx.y;
    const int l   = blockIdx.x * 256 + tid;

    for (int e = tid; e < u * 64; e += 256) {
        int s = e >> 6, d = e & 63;
        int idx = idx_sample[s];
        Ksamp[e] = K[((size_t)bh * L + idx) * 64 + d];
    }
    __syncthreads();

    float q[64];
    const float* Qp = Q + ((size_t)bh * L + l) * 64;
#pragma unroll
    for (int d = 0; d < 64; ++d) q[d] = Qp[d];

    float mx = -FLT_MAX, sm = 0.f;
    for (int s = 0; s < u; ++s) {
        float dot = 0.f;
#pragma unroll
        for (int d = 0; d < 64; ++d) dot += q[d] * Ksamp[(s << 6) + d];
        dot *= 0.125f;  // 1/sqrt(64)
        mx = fmaxf(mx, dot);
        sm += dot;
    }
    Msc[(size_t)bh * L + l] = mx - sm / (float)u;
}

// ---------------------------------------------------------------------------
// Exact top-u by repeated argmax over M (L=8192 fits in LDS). Lower-index wins ties.
// ---------------------------------------------------------------------------
__global__ __launch_bounds__(256) void topk_kernel(const float* __restrict__ Msc,
                                                   int* __restrict__ topidx,
                                                   int u, int L) {
    __shared__ float mlds[8192];
    __shared__ float sv[256];
    __shared__ int   si[256];
    const int tid = threadIdx.x;
    const int bh  = blockIdx.x;

    for (int i = tid; i < L; i += 256) mlds[i] = Msc[(size_t)bh * L + i];
    __syncthreads();

    for (int it = 0; it < u; ++it) {
        float bv = -FLT_MAX;
        int   bi = L;
        for (int i = tid; i < L; i += 256) {
            float v = mlds[i];
            if (v > bv || (v == bv && i < bi)) { bv = v; bi = i; }
        }
        sv[tid] = bv; si[tid] = bi;
        __syncthreads();
        for (int s2 = 128; s2 > 0; s2 >>= 1) {
            if (tid < s2) {
                if (sv[tid + s2] > sv[tid] ||
                    (sv[tid + s2] == sv[tid] && si[tid + s2] < si[tid])) {
                    sv[tid] = sv[tid + s2];
                    si[tid] = si[tid + s2];
                }
            }
            __syncthreads();
        }
        if (tid == 0) {
            topidx[bh * u + it] = si[0];
            mlds[si[0]] = -FLT_MAX;
        }
        __syncthreads();
    }
}

// ---------------------------------------------------------------------------
// Flash-style attention for the u top queries per (b,h): online softmax over L.
// ---------------------------------------------------------------------------
__global__ __launch_bounds__(256) void flash_kernel(const float* __restrict__ Q,
                                                    const float* __restrict__ K,
                                                    const float* __restrict__ V,
                                                    const int* __restrict__ topidx,
                                                    float* __restrict__ Vtop,
                                                    int u, int L) {
    __shared__ float Qs[MAXU * 64];
    __shared__ float Ks[TKF * 64];
    __shared__ float Vs[TKF * 64];
    __shared__ float Ss[MAXU * TKF];
    __shared__ float accS[MAXU * 64];
    __shared__ float row_m[MAXU], row_s[MAXU], row_f[MAXU];

    const int tid = threadIdx.x;
    const int bh  = blockIdx.x;
    const float scale = 0.125f;
    const float* Kb = K + (size_t)bh * L * 64;
    const float* Vb = V + (size_t)bh * L * 64;

    for (int e = tid; e < u * 64; e += 256) {
        int r = e >> 6, d = e & 63;
        int l = topidx[bh * u + r];
        Qs[e]   = Q[((size_t)bh * L + l) * 64 + d];
        accS[e] = 0.f;
    }
    if (tid < u) { row_m[tid] = -FLT_MAX; row_s[tid] = 0.f; }
    __syncthreads();

    for (int l0 = 0; l0 < L; l0 += TKF) {
        for (int e = tid; e < TKF * 64; e += 256) {
            Ks[e] = Kb[(size_t)l0 * 64 + e];
            Vs[e] = Vb[(size_t)l0 * 64 + e];
        }
        __syncthreads();
        for (int e = tid; e < u * TKF; e += 256) {
            int r = e / TKF, j = e % TKF;
            float dot = 0.f;
#pragma unroll
            for (int d = 0; d < 64; ++d) dot += Qs[r * 64 + d] * Ks[j * 64 + d];
            Ss[e] = dot * scale;
        }
        __syncthreads();
        if (tid < u) {
            float tm = -FLT_MAX;
            for (int j = 0; j < TKF; ++j) tm = fmaxf(tm, Ss[tid * TKF + j]);
            float nm  = fmaxf(row_m[tid], tm);
            row_f[tid] = __expf(row_m[tid] - nm);
            row_m[tid] = nm;
        }
        __syncthreads();
        for (int e = tid; e < u * TKF; e += 256) {
            int r = e / TKF;
            Ss[e] = __expf(Ss[e] - row_m[r]);
        }
        __syncthreads();
        if (tid < u) {
            float sm = 0.f;
            for (int j = 0; j < TKF; ++j) sm += Ss[tid * TKF + j];
            row_s[tid] = row_s[tid] * row_f[tid] + sm;
        }
        __syncthreads();
        for (int e = tid; e < u * 64; e += 256) {
            int r = e >> 6, d = e & 63;
            float a = accS[e] * row_f[r];
#pragma unroll
            for (int j = 0; j < TKF; ++j) a += Ss[r * TKF + j] * Vs[j * 64 + d];
            accS[e] = a;
        }
        __syncthreads();
    }

    for (int e = tid; e < u * 64; e += 256) {
        int r = e >> 6;
        Vtop[(size_t)bh * u * 64 + e] = accS[e] / row_s[r];
    }
}

// ---------------------------------------------------------------------------
// V mean over L per (b,h,d)
// ---------------------------------------------------------------------------
__global__ __launch_bounds__(256) void vmean_kernel(const float* __restrict__ V,
                                                    float* __restrict__ Vmean,
                                                    int L) {
    __shared__ float red[256];
    const int tid  = threadIdx.x;
    const int bh   = blockIdx.x;
    const int d    = tid & 63;
    const int part = tid >> 6;   // 4 partial sums per channel
    float s = 0.f;
    for (int l = part; l < L; l += 4)
        s += V[((size_t)bh * L + l) * 64 + d];
    red[tid] = s;
    __syncthreads();
    if (tid < 64) {
        float t = red[tid] + red[64 + tid] + red[128 + tid] + red[192 + tid];
        Vmean[bh * 64 + tid] = t / (float)L;
    }
}

// ---------------------------------------------------------------------------
// ctx[b,l,h*D+d] = Vmean[b,h,d]   (broadcast fill)
// ---------------------------------------------------------------------------
__global__ __launch_bounds__(256) void ctx_fill_kernel(float* __restrict__ ctx,
                                                       const float* __restrict__ Vmean,
                                                       int L, int H, int D, int dm) {
    size_t idx = (size_t)blockIdx.x * 256 + threadIdx.x;
    int col = (int)(idx % dm);
    size_t m = idx / dm;
    int b = (int)(m / L);
    int h = col >> 6, d = col & 63;
    ctx[idx] = Vmean[(b * H + h) * 64 + d];
}

// ---------------------------------------------------------------------------
// ctx[b, top_idx[b,h,r], h*D+d] = Vtop[b,h,r,d]
// ---------------------------------------------------------------------------
__global__ __launch_bounds__(256) void ctx_scatter_kernel(float* __restrict__ ctx,
                                                          const float* __restrict__ Vtop,
                                                          const int* __restrict__ topidx,
                                                          int u, int L, int H, int D,
                                                          int dm, int n) {
    int idx = blockIdx.x * 256 + threadIdx.x;
    if (idx >= n) return;
    int d  = idx & 63;
    int t  = idx >> 6;
    int r  = t % u;
    int bh = t / u;
    int b = bh / H, h = bh % H;
    int l = topidx[bh * u + r];
    ctx[((size_t)b * L + l) * dm + h * 64 + d] = Vtop[idx];
}

// ---------------------------------------------------------------------------
extern "C" void kernel_launch(void* const* d_in, const int* in_sizes, int n_in,
                              void* d_out, int out_size, void* d_ws, size_t ws_size,
                              hipStream_t stream) {
    const float* x   = (const float*)d_in[0];
    const float* Wq  = (const float*)d_in[1];
    const float* Wk  = (const float*)d_in[2];
    const float* Wv  = (const float*)d_in[3];
    const float* Wo  = (const float*)d_in[4];
    const int* idx_sample = (const int*)d_in[5];
    int u = in_sizes[5];
    if (u > MAXU) u = MAXU;   // LDS sizing guard (u == 45 at this problem size)

    const int B = 4, L = 8192, dm = 512, H = 8, D = 64;
    const int M = B * L;  // 32768

    float* ws    = (float*)d_ws;
    size_t nQ    = (size_t)M * dm;                     // 16,777,216 floats each
    float* Q     = ws;
    float* Kbuf  = ws + nQ;
    float* Vbuf  = ws + 2 * nQ;
    float* Msc   = ws + 3 * nQ;                        // B*H*L floats
    float* Vtop  = Msc + (size_t)B * H * L;            // B*H*u*D floats
    float* Vmean = Vtop + (size_t)B * H * u * D;       // B*H*D floats
    int*   topix = (int*)(Vmean + (size_t)B * H * D);  // B*H*u ints
    float* ctx   = Q;                                  // alias: Q dead after flash
    float* out   = (float*)d_out;

    dim3 gblk(dm / 64, M / 128);  // (8, 256)
    gemm_wmma<<<gblk, 256, 0, stream>>>(x, Wq, Q,    M, dm, dm, 1);
    gemm_wmma<<<gblk, 256, 0, stream>>>(x, Wk, Kbuf, M, dm, dm, 1);
    gemm_wmma<<<gblk, 256, 0, stream>>>(x, Wv, Vbuf, M, dm, dm, 1);

    sample_m_kernel<<<dim3(L / 256, B * H), 256, 0, stream>>>(Q, Kbuf, idx_sample, Msc, u, L);
    topk_kernel<<<B * H, 256, 0, stream>>>(Msc, topix, u, L);
    flash_kernel<<<B * H, 256, 0, stream>>>(Q, Kbuf, Vbuf, topix, Vtop, u, L);
    vmean_kernel<<<B * H, 256, 0, stream>>>(Vbuf, Vmean, L);

    ctx_fill_kernel<<<(M * dm) / 256, 256, 0, stream>>>(ctx, Vmean, L, H, D, dm);
    int nsc = B * H * u * D;
    ctx_scatter_kernel<<<(nsc + 255) / 256, 256, 0, stream>>>(ctx, Vtop, topix, u, L, H, D, dm, nsc);

    gemm_wmma<<<gblk, 256, 0, stream>>>(ctx, Wo, out, M, dm, dm, 0);
}